// DualOctreeGroupNorm_15487652069472
// MI455X (gfx1250) — compile-verified
//
#include <hip/hip_runtime.h>
#include <hip/hip_bf16.h>

// DualOctreeGroupNorm for MI455X (gfx1250, wave32).
// ws layout: floats [0,2048) channel sums, [2048,4096) channel sum-of-squares,
//            [4096,8192) affine table (scale[16*128], shift[16*128]),
//            ints    [8192*4 ..) segment starts[17].

constexpr int   kC   = 128;   // IN_CHANNELS
constexpr int   kB   = 16;    // batch segments
constexpr int   kCPG = 4;     // channels per group
constexpr float kEPS = 1e-5f;

typedef float v2f __attribute__((ext_vector_type(2)));
typedef float v4f __attribute__((ext_vector_type(4)));
typedef float v8f __attribute__((ext_vector_type(8)));

// ---------------- zero accumulators ----------------
__global__ void dogn_zero(float* __restrict__ p, int n) {
  int i = blockIdx.x * blockDim.x + threadIdx.x;
  if (i < n) p[i] = 0.0f;
}

// ---------------- segment starts (batch_id is sorted) ----------------
__global__ void dogn_starts(const int* __restrict__ bid, int* __restrict__ starts, int N) {
  const int t = threadIdx.x;
  if (t > kB) return;
  if (t == 0)       { starts[0]  = 0; return; }
  if (t == kB)      { starts[kB] = N; return; }
  int lo = 0, hi = N;                       // lower_bound(bid, t)
  while (lo < hi) { int mid = (lo + hi) >> 1; if (bid[mid] < t) lo = mid + 1; else hi = mid; }
  starts[t] = lo;
}

// ---------------- pass 1: segmented Σx / Σx² via WMMA ones-reduction ----------------
// Block = 256 threads = 8 waves; wave w owns channels [16w, 16w+16).
// B-matrix (4x16 f32): lanes 0-15 rows {r, r+1}, lanes 16-31 rows {r+2, r+3}.
// A = all-ones 16x4 -> every row of D accumulates the column sums of B.
// Segment bounds are precomputed, so the interior loop is pure load+mul+wmma.
__global__ void dogn_pass1(const float* __restrict__ data,
                           const int*   __restrict__ starts,
                           float* __restrict__ sums,
                           float* __restrict__ sqs,
                           int N, int rowsPerBlock)
{
  const int lane = threadIdx.x & 31;
  const int wave = threadIdx.x >> 5;
  const int ch   = (wave << 4) + (lane & 15);
  const int kk   = lane >> 4;

  const int rowBeg = blockIdx.x * rowsPerBlock;
  int rowEnd = rowBeg + rowsPerBlock;
  if (rowEnd > N) rowEnd = N;
  if (rowBeg >= rowEnd) return;

  const v2f ones = {1.0f, 1.0f};

  int s = 0;
  while (starts[s + 1] <= rowBeg) ++s;      // first overlapping segment (<=16 scalar-cached loads)

  for (; s < kB && starts[s] < rowEnd; ++s) {
    const int segBeg = max(starts[s], rowBeg);
    const int segEnd = min(starts[s + 1], rowEnd);
    const int len = segEnd - segBeg;
    if (len <= 0) continue;

    v8f c0e, c0o, c1e, c1o;                 // 4 independent WMMA chains (sum/sq x even/odd)
    #pragma unroll
    for (int i = 0; i < 8; ++i) { c0e[i] = 0.0f; c0o[i] = 0.0f; c1e[i] = 0.0f; c1o[i] = 0.0f; }

    const float* p = data + (size_t)(segBeg + 2 * kk) * kC + ch;
    const int nFull = len >> 2;             // full 4-row steps, no masking needed

    int i = 0;
    for (; i + 1 < nFull; i += 2) {         // 2 steps/iter: 4 clustered NT loads in flight
      const float a0 = __builtin_nontemporal_load(p);
      const float b0 = __builtin_nontemporal_load(p + kC);
      const float a1 = __builtin_nontemporal_load(p + 4 * kC);
      const float b1 = __builtin_nontemporal_load(p + 5 * kC);
      p += 8 * kC;
      v2f m0 = {a0, b0}, q0 = {a0 * a0, b0 * b0};
      v2f m1 = {a1, b1}, q1 = {a1 * a1, b1 * b1};
      c0e = __builtin_amdgcn_wmma_f32_16x16x4_f32(false, ones, false, m0, (short)0, c0e, false, false);
      c1e = __builtin_amdgcn_wmma_f32_16x16x4_f32(false, ones, false, q0, (short)0, c1e, false, false);
      c0o = __builtin_amdgcn_wmma_f32_16x16x4_f32(false, ones, false, m1, (short)0, c0o, false, false);
      c1o = __builtin_amdgcn_wmma_f32_16x16x4_f32(false, ones, false, q1, (short)0, c1o, false, false);
    }
    if (i < nFull) {                        // one leftover full step
      const float a0 = __builtin_nontemporal_load(p);
      const float b0 = __builtin_nontemporal_load(p + kC);
      p += 4 * kC;
      v2f m0 = {a0, b0}, q0 = {a0 * a0, b0 * b0};
      c0e = __builtin_amdgcn_wmma_f32_16x16x4_f32(false, ones, false, m0, (short)0, c0e, false, false);
      c1e = __builtin_amdgcn_wmma_f32_16x16x4_f32(false, ones, false, q0, (short)0, c1e, false, false);
    }

    const int rem = len & 3;                // 1..3 tail rows, masked by select (EXEC stays all-1)
    if (rem) {
      const int base = segBeg + (nFull << 2);
      const int ra = base + 2 * kk;
      const int rb = ra + 1;
      const int raC = min(ra, N - 1);
      const int rbC = min(rb, N - 1);
      const float xa = data[(size_t)raC * kC + ch];
      const float xb = data[(size_t)rbC * kC + ch];
      v2f m0, q0;
      m0.x = (ra < segEnd) ? xa : 0.0f;
      m0.y = (rb < segEnd) ? xb : 0.0f;
      q0.x = m0.x * m0.x;
      q0.y = m0.y * m0.y;
      c0e = __builtin_amdgcn_wmma_f32_16x16x4_f32(false, ones, false, m0, (short)0, c0e, false, false);
      c1e = __builtin_amdgcn_wmma_f32_16x16x4_f32(false, ones, false, q0, (short)0, c1e, false, false);
    }

    // D row M=0 (VGPR0, lanes 0-15) carries the column sums of each chain.
    if (lane < 16) {
      atomicAdd(&sums[s * kC + ch], c0e[0] + c0o[0]);
      atomicAdd(&sqs [s * kC + ch], c1e[0] + c1o[0]);
    }
  }
}

// ---------------- pass 2: per-(batch,channel) affine table ----------------
__global__ void dogn_pass2(const int*   __restrict__ starts,
                           const float* __restrict__ w,
                           const float* __restrict__ b,
                           const float* __restrict__ sums,
                           const float* __restrict__ sqs,
                           float* __restrict__ table)
{
  for (int idx = threadIdx.x; idx < kB * kC; idx += blockDim.x) {
    const int bb = idx / kC;
    const int c  = idx & (kC - 1);
    const int g0 = c & ~(kCPG - 1);
    const float* S = sums + bb * kC + g0;
    const float* Q = sqs  + bb * kC + g0;
    const float gs = S[0] + S[1] + S[2] + S[3];
    const float gq = Q[0] + Q[1] + Q[2] + Q[3];
    const float n  = (float)(starts[bb + 1] - starts[bb]);
    const float invc = 1.0f / (n * (float)kCPG + kEPS);          // ref: 1/(4n+eps)
    const float m    = gs * invc;                                 // group mean
    const float var  = (gq - 2.0f * m * gs + (float)kCPG * n * m * m) * invc;
    const float istd = rsqrtf(var + kEPS);
    const float sc   = istd * w[c];
    table[idx]           = sc;               // scale
    table[kB * kC + idx] = b[c] - m * sc;    // shift
  }
}

// ---------------- pass 3: out = x*scale[bid] + shift[bid] ----------------
// Table (16 KB) staged to LDS via CDNA5 async global->LDS loads; one wave = one 512B row.
__global__ void dogn_pass3(const float* __restrict__ data,
                           const int*   __restrict__ bid,
                           const float* __restrict__ table,
                           float* __restrict__ out, int N)
{
  __shared__ __align__(16) float tbl[2 * kB * kC];   // 4096 floats = 16 KB
  {
    const int tid = threadIdx.x;                     // 256 threads x 4 x b128 = 16 KB
    #pragma unroll
    for (int i = 0; i < 4; ++i) {
      const int e = tid * 4 + i * 1024;
      unsigned loff = (unsigned)(size_t)&tbl[e];     // generic LDS ptr: low 32 bits = LDS offset
      const float* g = table + e;
      asm volatile("global_load_async_to_lds_b128 %0, %1, off"
                   :: "v"(loff), "v"(g) : "memory");
    }
    asm volatile("s_wait_asynccnt 0" ::: "memory");
    __syncthreads();
  }

  const int lane = threadIdx.x & 31;
  const int wrow = threadIdx.x >> 5;                 // 8 rows per block-iter
  const int c4   = lane << 2;                        // 4 channels per lane
  long long row = (long long)blockIdx.x * 8 + wrow;
  const long long stride = (long long)gridDim.x * 8;
  #pragma unroll 4
  for (; row < N; row += stride) {
    const int s = bid[row];                          // broadcast within the wave
    const v4f x  = __builtin_nontemporal_load((const v4f*)(data + (size_t)row * kC + c4));
    const v4f sc = *(const v4f*)(&tbl[s * kC + c4]);
    const v4f sh = *(const v4f*)(&tbl[kB * kC + s * kC + c4]);
    const v4f o  = x * sc + sh;
    __builtin_nontemporal_store(o, (v4f*)(out + (size_t)row * kC + c4));
  }
}

extern "C" void kernel_launch(void* const* d_in, const int* in_sizes, int n_in,
                              void* d_out, int out_size, void* d_ws, size_t ws_size,
                              hipStream_t stream) {
  const float* data = (const float*)d_in[0];
  const float* w    = (const float*)d_in[1];
  const float* bia  = (const float*)d_in[2];
  const int*   bid  = (const int*)d_in[3];
  const int N = in_sizes[3];

  float* ws     = (float*)d_ws;
  float* sums   = ws;
  float* sqs    = ws + kB * kC;
  float* table  = ws + 2 * kB * kC;
  int*   starts = (int*)(ws + 4 * kB * kC);
  float* out    = (float*)d_out;

  dogn_zero<<<(2 * kB * kC + 255) / 256, 256, 0, stream>>>(ws, 2 * kB * kC);
  dogn_starts<<<1, 32, 0, stream>>>(bid, starts, N);

  const int rowsPerBlock = 1024;
  const int nb1 = (N + rowsPerBlock - 1) / rowsPerBlock;
  dogn_pass1<<<nb1, 256, 0, stream>>>(data, starts, sums, sqs, N, rowsPerBlock);

  dogn_pass2<<<1, 512, 0, stream>>>(starts, w, bia, sums, sqs, table);

  dogn_pass3<<<2048, 256, 0, stream>>>(data, bid, table, out, N);
}